// Model_14499809591703
// MI455X (gfx1250) — compile-verified
//
#include <hip/hip_runtime.h>
#include <math.h>

typedef float v2f __attribute__((ext_vector_type(2)));
typedef float v8f __attribute__((ext_vector_type(8)));
typedef unsigned int v4u __attribute__((ext_vector_type(4)));
typedef int v4i __attribute__((ext_vector_type(4)));
typedef int v8i __attribute__((ext_vector_type(8)));

#define NR   2048
#define DIN  128
#define HD   64
#define OUTD 7
#define KC   32     // K-chunk for the big A GEMM

__device__ __forceinline__ v8f wmma4(v2f a, v2f b, v8f c) {
  // V_WMMA_F32_16X16X4_F32: D = A(16x4 f32) * B(4x16 f32) + C(16x16 f32)
  return __builtin_amdgcn_wmma_f32_16x16x4_f32(
      /*neg_a=*/false, a, /*neg_b=*/false, b,
      /*c_mod=*/(short)0, c, /*reuse_a=*/false, /*reuse_b=*/false);
}

__device__ __forceinline__ float leaky_f(float x) { return x > 0.0f ? x : 0.1f * x; }

#if defined(__has_builtin)
#if __has_builtin(__builtin_amdgcn_tensor_load_to_lds) && __has_builtin(__builtin_amdgcn_s_wait_tensorcnt)
#define USE_TDM 1
#endif
#endif
#ifndef USE_TDM
#define USE_TDM 0
#endif

#if USE_TDM
// 2D tile DMA: global -> LDS via Tensor Data Mover.
// tdim*/til* in elements (4B), stride0 in elements. pad_interval encodes DWORDs
// between pads as log2-1 (4 => pad after every 32 DWORDs); pad_amount fixed = 1 DWORD.
__device__ __forceinline__ void tdm_load_2d(const void* gptr, unsigned lds_off,
                                            unsigned tdim0, unsigned tdim1,
                                            unsigned til0, unsigned til1,
                                            unsigned stride0, unsigned pad_en,
                                            unsigned pad_interval) {
  unsigned long long ga = (unsigned long long)gptr;
  v4u g0;
  g0.x = 1u;                                              // count=1, user descriptor
  g0.y = lds_off;                                         // LDS byte address
  g0.z = (unsigned)(ga & 0xFFFFFFFFu);                    // global_addr[31:0]
  g0.w = (unsigned)((ga >> 32) & 0x01FFFFFFu) | (2u << 30); // addr[56:32] | type=2
  v8i g1;
  g1[0] = (int)((2u << 16) | (pad_en << 20) | (pad_interval << 22)); // data_size=4B
  g1[1] = (int)(tdim0 << 16);                             // tensor_dim0[15:0]
  g1[2] = (int)((tdim0 >> 16) | (tdim1 << 16));           // dim0 hi | dim1 lo
  g1[3] = (int)((tdim1 >> 16) | (til0 << 16));            // dim1 hi | tile_dim0
  g1[4] = (int)til1;                                      // tile_dim1 | tile_dim2=0
  g1[5] = (int)stride0;                                   // tensor_dim0_stride[31:0]
  g1[6] = 0;
  g1[7] = 0;
  v4i z4 = {};
#if __clang_major__ >= 23
  v8i z8 = {};
  __builtin_amdgcn_tensor_load_to_lds(g0, g1, z4, z4, z8, 0);
#else
  __builtin_amdgcn_tensor_load_to_lds(g0, g1, z4, z4, 0);
#endif
}
#endif

// ---------------- workspace layout (float offsets) ----------------
enum : size_t {
  WS_D    = 0,                    // 2048  (A-laplacian d)
  WS_SQ   = WS_D    + NR,         // 2048  (row sq-norms of h)
  WS_DAF  = WS_SQ   + NR,         // 2048  (Af-laplacian d)
  WS_H    = WS_DAF  + NR,         // 2048x64
  WS_S    = WS_H    + NR*HD,      // 2048x64 (cos_sim vectors)
  WS_XD   = WS_S    + NR*HD,      // 2048x64 (d .* X staging for gemmA)
  WS_QT   = WS_XD   + NR*HD,      // At@h
  WS_ZT1  = WS_QT   + NR*HD,
  WS_ZCT1 = WS_ZT1  + NR*HD,
  WS_S1   = WS_ZCT1 + NR*HD,     // Af@h
  WS_ZF1  = WS_S1   + NR*HD,
  WS_ZCF1 = WS_ZF1  + NR*HD,
  WS_QTMP = WS_ZCF1 + NR*HD,
  WS_ZT   = WS_QTMP + NR*HD,
  WS_ZCT  = WS_ZT   + NR*HD,
  WS_ZF   = WS_ZCT  + NR*HD,
  WS_ZCF  = WS_ZF   + NR*HD,
  WS_LCF  = WS_ZCF  + NR*HD,
  WS_LCT  = WS_LCF  + NR*HD,
  WS_VAL4 = WS_LCT  + NR*HD,     // 2048*4 Af values
  WS_GFF  = WS_VAL4 + NR*4,      // 64x64 grams
  WS_GFT  = WS_GFF  + HD*HD,
  WS_GTT  = WS_GFT  + HD*HD,
  WS_C1   = WS_GTT  + HD*HD,
  WS_C2   = WS_C1   + HD*HD,
  WS_SZT  = WS_C2   + HD*HD,     // 4 x 64 column sums
  WS_SZCT = WS_SZT  + HD,
  WS_SZF  = WS_SZCT + HD,
  WS_SZCF = WS_SZF  + HD,
  WS_IDX4 = WS_SZCF + HD         // 2048*4 ints
};

// ---------------- d = rsqrt(rowsum(A)+1e-10) ----------------
__global__ void rowsum_d_kernel(const float* __restrict__ A, float* __restrict__ dvec) {
  __shared__ float red[256];
  int row = blockIdx.x;
  float s = 0.f;
  for (int c = threadIdx.x; c < NR; c += 256) s += A[row * NR + c];
  red[threadIdx.x] = s;
  __syncthreads();
  for (int st = 128; st > 0; st >>= 1) {
    if ((int)threadIdx.x < st) red[threadIdx.x] += red[threadIdx.x + st];
    __syncthreads();
  }
  if (threadIdx.x == 0) dvec[row] = rsqrtf(red[0] + 1e-10f);
}

// ---------------- h = leaky(x @ W1^T + b1)  [2048,128]x[128,64] ----------------
__global__ void gemm_h_kernel(const float* __restrict__ x, const float* __restrict__ W1,
                              const float* __restrict__ b1, float* __restrict__ h) {
  int lane = threadIdx.x & 31, wave = threadIdx.x >> 5;
  int i0 = (blockIdx.x * 8 + wave) * 16;
  int m = lane & 15, kb = (lane >> 4) * 2, hi = (lane >> 4) * 8;
  v8f acc[4] = {};
  for (int k = 0; k < DIN; k += 4) {
    v2f a;
    a.x = x[(i0 + m) * DIN + k + kb];
    a.y = x[(i0 + m) * DIN + k + kb + 1];
#pragma unroll
    for (int nt = 0; nt < 4; ++nt) {
      int n = nt * 16 + m;
      v2f b;
      b.x = W1[n * DIN + k + kb];
      b.y = W1[n * DIN + k + kb + 1];
      acc[nt] = wmma4(a, b, acc[nt]);
    }
  }
#pragma unroll
  for (int nt = 0; nt < 4; ++nt) {
    int n = nt * 16 + m;
    float bias = b1[n];
#pragma unroll
    for (int r = 0; r < 8; ++r)
      h[(i0 + r + hi) * HD + n] = leaky_f(acc[nt][r] + bias);
  }
}

// ---------------- per-row stats: sq = sum(h^2); s = h/sqrt(sq) + 1e-10 ----------------
__global__ void row_stats_kernel(const float* __restrict__ h, float* __restrict__ sq,
                                 float* __restrict__ s) {
  int row = blockIdx.x * blockDim.x + threadIdx.x;
  if (row >= NR) return;
  float acc = 0.f;
  for (int c = 0; c < HD; ++c) { float v = h[row * HD + c]; acc += v * v; }
  sq[row] = acc;
  float inv = rsqrtf(acc);
  for (int c = 0; c < HD; ++c) s[row * HD + c] = h[row * HD + c] * inv + 1e-10f;
}

// ---------------- fused heat-Gram + per-row top-4 (one wave / 16 rows) ----------------
__device__ __forceinline__ void top4_insert(float v, int j, float& v0, float& v1,
                                            float& v2_, float& v3, int& j0, int& j1,
                                            int& j2, int& j3) {
  if (v > v3) {
    if (v > v0)       { v3=v2_; j3=j2; v2_=v1; j2=j1; v1=v0; j1=j0; v0=v; j0=j; }
    else if (v > v1)  { v3=v2_; j3=j2; v2_=v1; j2=j1; v1=v;  j1=j; }
    else if (v > v2_) { v3=v2_; j3=j2; v2_=v;  j2=j; }
    else              { v3=v;   j3=j; }
  }
}

__global__ void topk_kernel(const float* __restrict__ h, const float* __restrict__ sq,
                            int* __restrict__ idx4) {
  __shared__ float ldsHi[16][65];
  __shared__ float ldsT[16][17];
  int lane = threadIdx.x;
  int i0 = blockIdx.x * 16;
  for (int idx = lane; idx < 16 * HD; idx += 32)
    ldsHi[idx >> 6][idx & 63] = h[(i0 + (idx >> 6)) * HD + (idx & 63)];
  __syncthreads();
  int m = lane & 15, kb = (lane >> 4) * 2, hi = (lane >> 4) * 8;
  float si[8];
#pragma unroll
  for (int r = 0; r < 8; ++r) si[r] = sq[i0 + r + hi];
  float bv0 = -1e30f, bv1 = -1e30f, bv2 = -1e30f, bv3 = -1e30f;
  int bj0 = 0, bj1 = 0, bj2 = 0, bj3 = 0;
  for (int j0t = 0; j0t < NR; j0t += 16) {
    v8f acc = {};
#pragma unroll
    for (int k = 0; k < HD; k += 4) {
      v2f a; a.x = ldsHi[m][k + kb]; a.y = ldsHi[m][k + kb + 1];
      v2f b; b.x = h[(j0t + m) * HD + k + kb]; b.y = h[(j0t + m) * HD + k + kb + 1];
      acc = wmma4(a, b, acc);
    }
    float sj = sq[j0t + m];
#pragma unroll
    for (int r = 0; r < 8; ++r) {
      float dist = si[r] + sj - 2.0f * acc[r] + 64.0f * 1e-10f;
      ldsT[r + hi][m] = expf(-0.5f * dist);
    }
    __syncthreads();
    if (lane < 16) {
      for (int c = 0; c < 16; ++c)
        top4_insert(ldsT[lane][c], j0t + c, bv0, bv1, bv2, bv3, bj0, bj1, bj2, bj3);
    }
    __syncthreads();
  }
  if (lane < 16) {
    int row = i0 + lane;
    idx4[row * 4 + 0] = bj0; idx4[row * 4 + 1] = bj1;
    idx4[row * 4 + 2] = bj2; idx4[row * 4 + 3] = bj3;
  }
}

// ---------------- Af sparse values: cos_sim at top-4 cols + laplacian d ----------------
__global__ void af_prep_kernel(const float* __restrict__ s, const int* __restrict__ idx4,
                               float* __restrict__ val4, float* __restrict__ dAf) {
  int row = blockIdx.x * blockDim.x + threadIdx.x;
  if (row >= NR) return;
  float rs = 0.f;
  for (int t = 0; t < 4; ++t) {
    int j = idx4[row * 4 + t];
    float dot = 0.f;
    for (int c = 0; c < HD; ++c) dot += s[row * HD + c] * s[j * HD + c];
    val4[row * 4 + t] = dot;
    rs += dot;
  }
  dAf[row] = rsqrtf(rs + 1e-10f);
}

__global__ void af_scale_kernel(float* __restrict__ val4, const int* __restrict__ idx4,
                                const float* __restrict__ dAf) {
  int row = blockIdx.x * blockDim.x + threadIdx.x;
  if (row >= NR) return;
  float di = dAf[row];
  for (int t = 0; t < 4; ++t)
    val4[row * 4 + t] *= di * dAf[idx4[row * 4 + t]];
}

// ---------------- Xd = d .* X (column-scale staging for gemmA) ----------------
__global__ void prescale_kernel(const float* __restrict__ X, const float* __restrict__ dvec,
                                float* __restrict__ Xd) {
  int gid = blockIdx.x * blockDim.x + threadIdx.x;
  if (gid >= NR * HD) return;
  Xd[gid] = X[gid] * dvec[gid >> 6];
}

// ---------------- Y = d[i] * (A @ Xd)   [2048,2048]x[2048,64] ----------------
// Double-buffered: A-tile (128xKC, LDS-padded stride 33) and Xd-tile (KCx64)
// streamed by the Tensor Data Mover, retired with s_wait_tensorcnt, WMMA overlapped.
__global__ void gemmA_kernel(const float* __restrict__ A, const float* __restrict__ Xd,
                             const float* __restrict__ dvec, float* __restrict__ Y) {
  __shared__ float ldsA[2][128 * (KC + 1)];
  __shared__ float ldsB[2][KC * 64];
  int tid = threadIdx.x, lane = tid & 31, wave = tid >> 5;
  int rowBlock = blockIdx.x * 128;
  int m = lane & 15, kb = (lane >> 4) * 2, hi = (lane >> 4) * 8;
  int r0 = wave * 16;
  v8f acc[4] = {};

#if USE_TDM
  if (tid == 0) {
    tdm_load_2d(A + (size_t)rowBlock * NR, (unsigned)(uintptr_t)&ldsA[0][0],
                KC, 128, KC, 128, NR, /*pad=*/1, /*interval: 32 DW=*/4);
    tdm_load_2d(Xd, (unsigned)(uintptr_t)&ldsB[0][0],
                HD, KC, HD, KC, HD, /*pad=*/0, 0);
  }
#endif
  int buf = 0;
  for (int kc = 0; kc < NR; kc += KC, buf ^= 1) {
#if USE_TDM
    if (kc + KC < NR) {
      if (tid == 0) {
        tdm_load_2d(A + (size_t)rowBlock * NR + (kc + KC),
                    (unsigned)(uintptr_t)&ldsA[buf ^ 1][0],
                    KC, 128, KC, 128, NR, 1, 4);
        tdm_load_2d(Xd + (size_t)(kc + KC) * HD,
                    (unsigned)(uintptr_t)&ldsB[buf ^ 1][0],
                    HD, KC, HD, KC, HD, 0, 0);
      }
      __builtin_amdgcn_s_wait_tensorcnt(2);   // current chunk's two tiles retired
    } else {
      __builtin_amdgcn_s_wait_tensorcnt(0);
    }
#else
    // fallback: manual VGPR staging of the current chunk
    __syncthreads();
    for (int idx = tid; idx < 128 * KC; idx += 256) {
      int r = idx / KC, k = idx % KC;
      ldsA[buf][r * (KC + 1) + k] = A[(size_t)(rowBlock + r) * NR + kc + k];
    }
    for (int idx = tid; idx < KC * 64; idx += 256) {
      int k = idx >> 6, n = idx & 63;
      ldsB[buf][k * 64 + n] = Xd[(size_t)(kc + k) * HD + n];
    }
#endif
    __syncthreads();
    const float* lA = &ldsA[buf][0];
    const float* lB = &ldsB[buf][0];
#pragma unroll
    for (int kk = 0; kk < KC; kk += 4) {
      v2f a;
      a.x = lA[(r0 + m) * (KC + 1) + kk + kb];
      a.y = lA[(r0 + m) * (KC + 1) + kk + kb + 1];
#pragma unroll
      for (int nt = 0; nt < 4; ++nt) {
        v2f b;
        b.x = lB[(kk + kb) * 64 + nt * 16 + m];
        b.y = lB[(kk + kb + 1) * 64 + nt * 16 + m];
        acc[nt] = wmma4(a, b, acc[nt]);
      }
    }
    __syncthreads();
  }
#pragma unroll
  for (int nt = 0; nt < 4; ++nt)
#pragma unroll
    for (int r = 0; r < 8; ++r) {
      int row = rowBlock + r0 + r + hi;
      Y[row * HD + nt * 16 + m] = acc[nt][r] * dvec[row];
    }
}

// ---------------- Z = leaky(Y @ W^T + b)  [2048,64]x[64,64] ----------------
__global__ void gemmW_kernel(const float* __restrict__ Yin, const float* __restrict__ W,
                             const float* __restrict__ bias, float* __restrict__ Z) {
  __shared__ float ldsB[64][65];   // ldsB[k][n] = W[n][k]
  int tid = threadIdx.x, lane = tid & 31, wave = tid >> 5;
  for (int idx = tid; idx < HD * HD; idx += 256) {
    int n = idx >> 6, k = idx & 63;
    ldsB[k][n] = W[n * HD + k];
  }
  __syncthreads();
  int i0 = (blockIdx.x * 8 + wave) * 16;
  int m = lane & 15, kb = (lane >> 4) * 2, hi = (lane >> 4) * 8;
  v8f acc[4] = {};
#pragma unroll
  for (int k = 0; k < HD; k += 4) {
    v2f a;
    a.x = Yin[(i0 + m) * HD + k + kb];
    a.y = Yin[(i0 + m) * HD + k + kb + 1];
#pragma unroll
    for (int nt = 0; nt < 4; ++nt) {
      v2f b; b.x = ldsB[k + kb][nt * 16 + m]; b.y = ldsB[k + kb + 1][nt * 16 + m];
      acc[nt] = wmma4(a, b, acc[nt]);
    }
  }
#pragma unroll
  for (int nt = 0; nt < 4; ++nt) {
    int n = nt * 16 + m;
    float bv = bias[n];
#pragma unroll
    for (int r = 0; r < 8; ++r)
      Z[(i0 + r + hi) * HD + n] = leaky_f(acc[nt][r] + bv);
  }
}

// ---------------- sparse Af @ X (4 nnz per row) ----------------
__global__ void spmm_kernel(const float* __restrict__ val4, const int* __restrict__ idx4,
                            const float* __restrict__ X, float* __restrict__ Y) {
  int gid = blockIdx.x * blockDim.x + threadIdx.x;
  if (gid >= NR * HD) return;
  int i = gid >> 6, n = gid & 63;
  float acc = 0.f;
#pragma unroll
  for (int t = 0; t < 4; ++t)
    acc += val4[i * 4 + t] * X[idx4[i * 4 + t] * HD + n];
  Y[gid] = acc;
}

// ---------------- L = z / (sqrt(mean(z^2)) + 1e-10) ----------------
__global__ void rownorm_kernel(const float* __restrict__ z, float* __restrict__ L) {
  int row = blockIdx.x * blockDim.x + threadIdx.x;
  if (row >= NR) return;
  float acc = 0.f;
  for (int c = 0; c < HD; ++c) { float v = z[row * HD + c]; acc += v * v; }
  float inv = 1.0f / (sqrtf(acc * (1.0f / HD)) + 1e-10f);
  for (int c = 0; c < HD; ++c) L[row * HD + c] = z[row * HD + c] * inv;
}

// ---------------- G[a][b] = sum_i X[i][a]*Y[i][b]  (64x64, 16 waves) ----------------
__global__ void gram_kernel(const float* __restrict__ X, const float* __restrict__ Y,
                            float* __restrict__ G) {
  int lane = threadIdx.x & 31, wave = threadIdx.x >> 5;
  int m0 = (wave >> 2) * 16, n0 = (wave & 3) * 16;
  int m = lane & 15, kb = (lane >> 4) * 2, hi = (lane >> 4) * 8;
  v8f acc = {};
  for (int k = 0; k < NR; k += 4) {
    v2f a; a.x = X[(k + kb) * HD + m0 + m]; a.y = X[(k + kb + 1) * HD + m0 + m];
    v2f b; b.x = Y[(k + kb) * HD + n0 + m]; b.y = Y[(k + kb + 1) * HD + n0 + m];
    acc = wmma4(a, b, acc);
  }
#pragma unroll
  for (int r = 0; r < 8; ++r)
    G[(m0 + r + hi) * HD + n0 + m] = acc[r];
}

// ---------------- column sums ----------------
__global__ void colsum_kernel(const float* __restrict__ X, float* __restrict__ s) {
  int c = threadIdx.x;
  float acc = 0.f;
  for (int i = 0; i < NR; ++i) acc += X[i * HD + c];
  s[c] = acc;
}

// ---------------- Lc = (||Gff||^2 - 2||Gft||^2 + ||Gtt||^2) / N^2 ----------------
__global__ void lc_final_kernel(const float* __restrict__ Gff, const float* __restrict__ Gft,
                                const float* __restrict__ Gtt, float* __restrict__ out) {
  __shared__ float red[256];
  float acc = 0.f;
  for (int e = threadIdx.x; e < HD * HD; e += 256) {
    float a = Gff[e], b = Gft[e], c = Gtt[e];
    acc += a * a - 2.0f * b * b + c * c;
  }
  red[threadIdx.x] = acc;
  __syncthreads();
  for (int st = 128; st > 0; st >>= 1) {
    if ((int)threadIdx.x < st) red[threadIdx.x] += red[threadIdx.x + st];
    __syncthreads();
  }
  if (threadIdx.x == 0) out[0] = red[0] / ((float)NR * (float)NR);
}

// ---------------- Ld = (||C1 - szct szt^T/n||^2 + ||C2 - szcf szf^T/n||^2)/(n-1)^2 ---
__global__ void ld_final_kernel(const float* __restrict__ C1, const float* __restrict__ szct,
                                const float* __restrict__ szt, const float* __restrict__ C2,
                                const float* __restrict__ szcf, const float* __restrict__ szf,
                                float* __restrict__ out) {
  __shared__ float red[256];
  const float invn = 1.0f / (float)NR;
  float acc = 0.f;
  for (int e = threadIdx.x; e < HD * HD; e += 256) {
    int a = e >> 6, b = e & 63;
    float m1 = C1[e] - szct[a] * szt[b] * invn;
    float m2 = C2[e] - szcf[a] * szf[b] * invn;
    acc += m1 * m1 + m2 * m2;
  }
  red[threadIdx.x] = acc;
  __syncthreads();
  for (int st = 128; st > 0; st >>= 1) {
    if ((int)threadIdx.x < st) red[threadIdx.x] += red[threadIdx.x + st];
    __syncthreads();
  }
  if (threadIdx.x == 0) out[0] = red[0] / ((float)(NR - 1) * (float)(NR - 1));
}

// ---------------- attention aggregate + output head (one block / row) ----------------
__global__ void attn_out_kernel(const float* __restrict__ zf, const float* __restrict__ zt,
                                const float* __restrict__ zcf, const float* __restrict__ zct,
                                const float* __restrict__ Wa, const float* __restrict__ ba,
                                const float* __restrict__ q, const float* __restrict__ W2,
                                const float* __restrict__ b2, float* __restrict__ out) {
  __shared__ float z[3][HD];
  __shared__ float red[HD];
  __shared__ float aw[3];
  __shared__ float zag[HD];
  int row = blockIdx.x, t = threadIdx.x;
  z[0][t] = zf[row * HD + t];
  z[1][t] = zt[row * HD + t];
  z[2][t] = 0.5f * (zcf[row * HD + t] + zct[row * HD + t]);
  __syncthreads();
  for (int b = 0; b < 3; ++b) {
    float u = ba[t];
    for (int k = 0; k < HD; ++k) u += Wa[t * HD + k] * z[b][k];
    red[t] = tanhf(u) * q[t];
    __syncthreads();
    for (int st = 32; st > 0; st >>= 1) {
      if (t < st) red[t] += red[t + st];
      __syncthreads();
    }
    if (t == 0) aw[b] = red[0];
    __syncthreads();
  }
  if (t == 0) {
    float mx = fmaxf(aw[0], fmaxf(aw[1], aw[2]));
    float e0 = expf(aw[0] - mx), e1 = expf(aw[1] - mx), e2 = expf(aw[2] - mx);
    float inv = 1.0f / (e0 + e1 + e2);
    aw[0] = e0 * inv; aw[1] = e1 * inv; aw[2] = e2 * inv;
  }
  __syncthreads();
  zag[t] = aw[0] * z[0][t] + aw[1] * z[1][t] + aw[2] * z[2][t];
  __syncthreads();
  if (t < OUTD) {
    float o = b2[t];
    for (int k = 0; k < HD; ++k) o += W2[t * HD + k] * zag[k];
    out[row * OUTD + t] = o;
  }
}

// =====================================================================
extern "C" void kernel_launch(void* const* d_in, const int* in_sizes, int n_in,
                              void* d_out, int out_size, void* d_ws, size_t ws_size,
                              hipStream_t stream) {
  (void)in_sizes; (void)n_in; (void)out_size; (void)ws_size;
  const float* x   = (const float*)d_in[0];
  const float* A   = (const float*)d_in[1];
  /* d_in[2] = k (==4, hardcoded in topk path) */
  const float* W1  = (const float*)d_in[3];
  const float* b1  = (const float*)d_in[4];
  const float* WF1 = (const float*)d_in[5];
  const float* bF1 = (const float*)d_in[6];
  const float* WF2 = (const float*)d_in[7];
  const float* bF2 = (const float*)d_in[8];
  const float* WT1 = (const float*)d_in[9];
  const float* bT1 = (const float*)d_in[10];
  const float* WT2 = (const float*)d_in[11];
  const float* bT2 = (const float*)d_in[12];
  const float* WC1 = (const float*)d_in[13];
  const float* bC1 = (const float*)d_in[14];
  const float* WC2 = (const float*)d_in[15];
  const float* bC2 = (const float*)d_in[16];
  const float* Wa  = (const float*)d_in[17];
  const float* ba  = (const float*)d_in[18];
  const float* q   = (const float*)d_in[19];
  const float* W2  = (const float*)d_in[20];
  const float* b2  = (const float*)d_in[21];

  float* ws = (float*)d_ws;
  float* out = (float*)d_out;
  float* dvec = ws + WS_D;   float* sq   = ws + WS_SQ;  float* dAf  = ws + WS_DAF;
  float* h    = ws + WS_H;   float* s    = ws + WS_S;   float* Xd   = ws + WS_XD;
  float* Qt   = ws + WS_QT;
  float* zt1  = ws + WS_ZT1; float* zct1 = ws + WS_ZCT1;
  float* S1   = ws + WS_S1;  float* zf1  = ws + WS_ZF1; float* zcf1 = ws + WS_ZCF1;
  float* Qtmp = ws + WS_QTMP;
  float* zt   = ws + WS_ZT;  float* zct  = ws + WS_ZCT;
  float* zf   = ws + WS_ZF;  float* zcf  = ws + WS_ZCF;
  float* Lcf  = ws + WS_LCF; float* Lct  = ws + WS_LCT;
  float* val4 = ws + WS_VAL4;
  float* Gff  = ws + WS_GFF; float* Gft  = ws + WS_GFT; float* Gtt = ws + WS_GTT;
  float* C1   = ws + WS_C1;  float* C2   = ws + WS_C2;
  float* szt  = ws + WS_SZT; float* szct = ws + WS_SZCT;
  float* szf  = ws + WS_SZF; float* szcf = ws + WS_SZCF;
  int*   idx4 = (int*)(ws + WS_IDX4);

  // graph laplacian of A
  rowsum_d_kernel<<<NR, 256, 0, stream>>>(A, dvec);
  // h = leaky(x W1^T + b1)
  gemm_h_kernel<<<NR / 128, 256, 0, stream>>>(x, W1, b1, h);
  row_stats_kernel<<<NR / 256, 256, 0, stream>>>(h, sq, s);
  // fused heat Gram + top-4  -> Af sparse structure
  topk_kernel<<<NR / 16, 32, 0, stream>>>(h, sq, idx4);
  af_prep_kernel<<<NR / 256, 256, 0, stream>>>(s, idx4, val4, dAf);
  af_scale_kernel<<<NR / 256, 256, 0, stream>>>(val4, idx4, dAf);
  // shared first-layer propagations
  prescale_kernel<<<NR * HD / 256, 256, 0, stream>>>(h, dvec, Xd);
  gemmA_kernel<<<NR / 128, 256, 0, stream>>>(A, Xd, dvec, Qt);        // At@h
  gemmW_kernel<<<NR / 128, 256, 0, stream>>>(Qt, WT1, bT1, zt1);
  gemmW_kernel<<<NR / 128, 256, 0, stream>>>(Qt, WC1, bC1, zct1);
  spmm_kernel<<<NR * HD / 256, 256, 0, stream>>>(val4, idx4, h, S1);  // Af@h
  gemmW_kernel<<<NR / 128, 256, 0, stream>>>(S1, WF1, bF1, zf1);
  gemmW_kernel<<<NR / 128, 256, 0, stream>>>(S1, WC1, bC1, zcf1);
  // second layers
  prescale_kernel<<<NR * HD / 256, 256, 0, stream>>>(zt1, dvec, Xd);
  gemmA_kernel<<<NR / 128, 256, 0, stream>>>(A, Xd, dvec, Qtmp);
  gemmW_kernel<<<NR / 128, 256, 0, stream>>>(Qtmp, WT2, bT2, zt);
  prescale_kernel<<<NR * HD / 256, 256, 0, stream>>>(zct1, dvec, Xd);
  gemmA_kernel<<<NR / 128, 256, 0, stream>>>(A, Xd, dvec, Qtmp);
  gemmW_kernel<<<NR / 128, 256, 0, stream>>>(Qtmp, WC2, bC2, zct);
  spmm_kernel<<<NR * HD / 256, 256, 0, stream>>>(val4, idx4, zf1, Qtmp);
  gemmW_kernel<<<NR / 128, 256, 0, stream>>>(Qtmp, WF2, bF2, zf);
  spmm_kernel<<<NR * HD / 256, 256, 0, stream>>>(val4, idx4, zcf1, Qtmp);
  gemmW_kernel<<<NR / 128, 256, 0, stream>>>(Qtmp, WC2, bC2, zcf);
  // Lc via 64x64 grams of row-normalized embeddings
  rownorm_kernel<<<NR / 256, 256, 0, stream>>>(zcf, Lcf);
  rownorm_kernel<<<NR / 256, 256, 0, stream>>>(zct, Lct);
  gram_kernel<<<1, 512, 0, stream>>>(Lcf, Lcf, Gff);
  gram_kernel<<<1, 512, 0, stream>>>(Lcf, Lct, Gft);
  gram_kernel<<<1, 512, 0, stream>>>(Lct, Lct, Gtt);
  // Ld (HSIC) via centered cross-grams
  gram_kernel<<<1, 512, 0, stream>>>(zct, zt, C1);
  gram_kernel<<<1, 512, 0, stream>>>(zcf, zf, C2);
  colsum_kernel<<<1, HD, 0, stream>>>(zt, szt);
  colsum_kernel<<<1, HD, 0, stream>>>(zct, szct);
  colsum_kernel<<<1, HD, 0, stream>>>(zf, szf);
  colsum_kernel<<<1, HD, 0, stream>>>(zcf, szcf);
  lc_final_kernel<<<1, 256, 0, stream>>>(Gff, Gft, Gtt, out + NR * OUTD);
  ld_final_kernel<<<1, 256, 0, stream>>>(C1, szct, szt, C2, szcf, szf, out + NR * OUTD + 1);
  // attention aggregation + output head
  attn_out_kernel<<<NR, HD, 0, stream>>>(zf, zt, zcf, zct, Wa, ba, q, W2, b2, out);
}